// Green_22643067584728
// MI455X (gfx1250) — compile-verified
//
#include <hip/hip_runtime.h>
#include <hip/hip_bf16.h>
#include <math.h>
#include <stdint.h>

typedef __attribute__((ext_vector_type(2))) float v2f;
typedef __attribute__((ext_vector_type(8))) float v8f;

#define B_    32
#define C_    64
#define T_    2500
#define SFREQ_ 250.0f
#define K_    125
#define F_    10
#define TC_   (T_ - K_ + 1)            /* 2376 */
#define VEC_  2080                     /* C*(C+1)/2 */
#define FEAT_ (F_ * VEC_)              /* 20800 */
#define HID_  8
#define NOUT_ 2
#define EPS_  1e-5f
#define PI_   3.14159265358979f

#define TILE_ 32                       /* time tile for fused conv+cov */
#define XW_   (TILE_ + K_ - 1)         /* 156 (multiple of 4 -> clean B128 groups) */
#define NT_   ((TC_ + TILE_ - 1) / TILE_)  /* 75 */
#define ZP_   33                       /* Z tile pitch (odd -> no bank conflicts) */
#define AP_   65                       /* eigensolver LDS pitch */
#define NG4_  ((C_ * XW_) / 4)         /* 2496 float4 groups per X tile */
#define NIT_  10                       /* ceil(2496/256): uniform async issues/wave */

// ---------------------------------------------------------------------------
// K1: build Morlet kernels: kc[f][k] = env*cos(phase), ks = env*sin(phase),
// env normalized to unit sum per f.  10 blocks x 128 threads.
// ---------------------------------------------------------------------------
__global__ void k_build(const float* __restrict__ foi,
                        const float* __restrict__ fwhm,
                        float* __restrict__ kc, float* __restrict__ ks) {
  const int f = blockIdx.x;
  const int k = threadIdx.x;
  __shared__ float ssum[128];
  float e = 0.f, t = 0.f;
  if (k < K_) {
    t = ((float)k - (K_ - 1) * 0.5f) / SFREQ_;
    float fw = exp2f(fwhm[f]);
    float sigma = fw / (2.f * sqrtf(2.f * logf(2.f)));
    float z = t / sigma;
    e = expf(-0.5f * z * z);
  }
  ssum[k] = e;
  __syncthreads();
  for (int s = 64; s > 0; s >>= 1) {
    if (k < s) ssum[k] += ssum[k + s];
    __syncthreads();
  }
  const float inv = 1.f / ssum[0];
  if (k < K_) {
    float fhz = exp2f(foi[f]);
    float ph = 2.f * PI_ * fhz * t;
    kc[f * K_ + k] = e * inv * cosf(ph);
    ks[f * K_ + k] = e * inv * sinf(ph);
  }
}

// ---------------------------------------------------------------------------
// Async DMA of one X time-tile (64ch x 156 samples) into an LDS buffer using
// GLOBAL_LOAD_ASYNC_TO_LDS_B128 (GVS mode).  Exactly NIT_ instructions per
// wave (indices clamped; duplicate same-data LDS writes are benign) so the
// caller can retire a whole tile with a constant s_wait_asynccnt.
// Global offsets clamped in-bounds; t >= T entries are zero-scrubbed later.
// ---------------------------------------------------------------------------
__device__ __forceinline__ void issue_tile_async(const float* __restrict__ Xb,
                                                 const float* XsBuf, int t0,
                                                 int tid) {
  const uint32_t ldsbase = (uint32_t)(uintptr_t)XsBuf;  // low 32 bits = LDS offset
#pragma unroll
  for (int it = 0; it < NIT_; ++it) {
    int idx4 = tid + it * 256;
    if (idx4 > NG4_ - 1) idx4 = NG4_ - 1;     // uniform per-wave issue count
    const int idx = idx4 * 4;
    const int c = idx / XW_;
    const int off = idx % XW_;
    int tg = t0 + off;
    if (tg > T_ - 4) tg = T_ - 4;             // stay in-bounds (last tile only)
    const uint32_t voff = (uint32_t)((c * T_ + tg) * 4);
    const uint32_t laddr = ldsbase + (uint32_t)(idx * 4);
    asm volatile("global_load_async_to_lds_b128 %0, %1, %2 offset:0"
                 :
                 : "v"(laddr), "v"(voff), "s"(Xb)
                 : "memory");
  }
}

// ---------------------------------------------------------------------------
// K2: fused conv + covariance SYRK.  One block per (b,f) = 320 blocks,
// 256 threads = 8 waves.  Double-buffered async X staging overlaps the DMA of
// tile t+1 with conv + WMMA of tile t.  Covariance accumulates with
// V_WMMA_F32_16X16X4_F32 (2 of the 16 output tiles per wave).
// Writes Craw = Ccov / Tc.
// ---------------------------------------------------------------------------
__global__ __launch_bounds__(256) void k_conv_cov(
    const float* __restrict__ X, const float* __restrict__ kc,
    const float* __restrict__ ks, float* __restrict__ Craw) {
  const int g = blockIdx.x;                 // b*F + f
  const int b = g / F_, f = g % F_;
  const int tid = threadIdx.x;
  const int lane = tid & 31, wave = tid >> 5;

  __shared__ float Xs[2][C_ * XW_];         // 2 x 39.9 KB (double buffer)
  __shared__ float Zr[C_ * ZP_];            // 8.4 KB
  __shared__ float Zi[C_ * ZP_];            // 8.4 KB
  __shared__ float kcs[K_], kss[K_];

  if (tid < K_) { kcs[tid] = kc[f * K_ + tid]; kss[tid] = ks[f * K_ + tid]; }

  // two 16x16 output tiles per wave (16 tiles total over 8 waves)
  const int p0 = wave * 2, p1 = wave * 2 + 1;
  const int ti0 = p0 >> 2, tj0 = p0 & 3;
  const int ti1 = p1 >> 2, tj1 = p1 & 3;
  v8f acc0 = {};
  v8f acc1 = {};

  const float* Xb = X + (size_t)b * C_ * T_;
  const int tt = tid & 31;                  // time-in-tile
  const int cg = tid >> 5;                  // channel group (8 channels each)

  // WMMA fragment addressing (A 16x4 f32: lanes 0-15 K={0,1}, 16-31 K={2,3})
  const int mrow = lane & 15;
  const int koff = (lane < 16) ? 0 : 2;

  // prologue: DMA tile 0 into buffer 0
  issue_tile_async(Xb, Xs[0], 0, tid);

  int cur = 0;
  for (int tile = 0; tile < NT_; ++tile) {
    const int t0 = tile * TILE_;
    if (tile + 1 < NT_) {
      // DMA next tile into the other buffer, then retire only the previous
      // tile's NIT_ in-order async loads (next tile's stay in flight).
      issue_tile_async(Xb, Xs[1 - cur], t0 + TILE_, tid);
      asm volatile("s_wait_asynccnt 0xa" ::: "memory");
    } else {
      asm volatile("s_wait_asynccnt 0x0" ::: "memory");
    }
    __syncthreads();                        // tile data visible; Z reuse safe

    if (t0 + XW_ > T_) {                    // final tile only: scrub t >= T
      for (int idx = tid; idx < C_ * XW_; idx += 256) {
        int off = idx % XW_;
        if (t0 + off >= T_) Xs[cur][idx] = 0.f;
      }
      __syncthreads();
    }

    {
      float ar[8] = {0, 0, 0, 0, 0, 0, 0, 0};
      float ai[8] = {0, 0, 0, 0, 0, 0, 0, 0};
      const float* Xc = Xs[cur];
      for (int k = 0; k < K_; ++k) {
        const float wc = kcs[k], ws = kss[k];
#pragma unroll
        for (int u = 0; u < 8; ++u) {
          float x = Xc[(cg * 8 + u) * XW_ + tt + k];
          ar[u] = fmaf(x, wc, ar[u]);
          ai[u] = fmaf(x, ws, ai[u]);
        }
      }
      const bool valid = (t0 + tt) < TC_;   // zero-pad past Tc (no cov effect)
#pragma unroll
      for (int u = 0; u < 8; ++u) {
        Zr[(cg * 8 + u) * ZP_ + tt] = valid ? ar[u] : 0.f;
        Zi[(cg * 8 + u) * ZP_ + tt] = valid ? ai[u] : 0.f;
      }
    }
    __syncthreads();
    // SYRK accumulate: C[m][n] += sum_k Z[ti*16+m][k] * Z[tj*16+n][k]
    for (int k0 = 0; k0 < TILE_; k0 += 4) {
      const int kk = k0 + koff;
      v2f a0 = {Zr[(ti0 * 16 + mrow) * ZP_ + kk], Zr[(ti0 * 16 + mrow) * ZP_ + kk + 1]};
      v2f b0 = {Zr[(tj0 * 16 + mrow) * ZP_ + kk], Zr[(tj0 * 16 + mrow) * ZP_ + kk + 1]};
      acc0 = __builtin_amdgcn_wmma_f32_16x16x4_f32(false, a0, false, b0, (short)0, acc0, false, false);
      v2f a0i = {Zi[(ti0 * 16 + mrow) * ZP_ + kk], Zi[(ti0 * 16 + mrow) * ZP_ + kk + 1]};
      v2f b0i = {Zi[(tj0 * 16 + mrow) * ZP_ + kk], Zi[(tj0 * 16 + mrow) * ZP_ + kk + 1]};
      acc0 = __builtin_amdgcn_wmma_f32_16x16x4_f32(false, a0i, false, b0i, (short)0, acc0, false, false);
      v2f a1 = {Zr[(ti1 * 16 + mrow) * ZP_ + kk], Zr[(ti1 * 16 + mrow) * ZP_ + kk + 1]};
      v2f b1 = {Zr[(tj1 * 16 + mrow) * ZP_ + kk], Zr[(tj1 * 16 + mrow) * ZP_ + kk + 1]};
      acc1 = __builtin_amdgcn_wmma_f32_16x16x4_f32(false, a1, false, b1, (short)0, acc1, false, false);
      v2f a1i = {Zi[(ti1 * 16 + mrow) * ZP_ + kk], Zi[(ti1 * 16 + mrow) * ZP_ + kk + 1]};
      v2f b1i = {Zi[(tj1 * 16 + mrow) * ZP_ + kk], Zi[(tj1 * 16 + mrow) * ZP_ + kk + 1]};
      acc1 = __builtin_amdgcn_wmma_f32_16x16x4_f32(false, a1i, false, b1i, (short)0, acc1, false, false);
    }
    cur = 1 - cur;
  }
  // C/D layout: VGPR r, lanes 0-15 -> M=r, lanes 16-31 -> M=r+8; N = lane%16
  const float invT = 1.f / (float)TC_;
  float* Cg = Craw + (size_t)g * C_ * C_;
  const int n = lane & 15;
  const int mbase = (lane < 16) ? 0 : 8;
#pragma unroll
  for (int r = 0; r < 8; ++r) {
    Cg[(ti0 * 16 + mbase + r) * C_ + tj0 * 16 + n] = acc0[r] * invT;
    Cg[(ti1 * 16 + mbase + r) * C_ + tj1 * 16 + n] = acc1[r] * invT;
  }
}

// ---------------------------------------------------------------------------
// K3: shrinkage + parallel cyclic Jacobi eigensolver (64x64) + matrix log
// reconstruction logC = (V diag(log w)) V^T via WMMA + upper-tri vectorize.
// One block of 64 threads (2 waves) per (b,f) matrix; 320 blocks.
// ---------------------------------------------------------------------------
__global__ __launch_bounds__(64) void k_eig_logm(
    const float* __restrict__ Craw, const float* __restrict__ shrink,
    float* __restrict__ vecbuf) {
  const int g = blockIdx.x;
  const int tid = threadIdx.x;
  const int lane = tid & 31, wave = tid >> 5;

  __shared__ float A[C_ * AP_];   // 16.6 KB, pitch 65 avoids bank conflicts
  __shared__ float V[C_ * AP_];   // 16.6 KB
  __shared__ float lw[C_];
  __shared__ float red[C_];
  __shared__ float rc[32], rs[32];
  __shared__ int pp[32], qq[32], topi[32], boti[32];

  const float* Cg = Craw + (size_t)g * C_ * C_;
  for (int idx = tid; idx < C_ * C_; idx += 64) {
    int i = idx >> 6, j = idx & 63;
    A[i * AP_ + j] = Cg[idx];
    V[i * AP_ + j] = (i == j) ? 1.f : 0.f;
  }
  __syncthreads();
  red[tid] = A[tid * AP_ + tid];
  __syncthreads();
  for (int s = 32; s > 0; s >>= 1) {
    if (tid < s) red[tid] += red[tid + s];
    __syncthreads();
  }
  const float mu = red[0] * (1.f / 64.f);
  const float alpha = 1.f / (1.f + expf(-shrink[0]));
  for (int idx = tid; idx < C_ * C_; idx += 64) {
    int i = idx >> 6, j = idx & 63;
    float v = A[i * AP_ + j] * (1.f - alpha);
    if (i == j) v += alpha * mu;
    A[i * AP_ + j] = v;
  }
  if (tid < 32) { topi[tid] = 2 * tid; boti[tid] = 2 * tid + 1; }
  __syncthreads();

  // cyclic Jacobi: 10 sweeps x 63 round-robin rounds of 32 disjoint rotations
  for (int sweep = 0; sweep < 10; ++sweep) {
    for (int round = 0; round < 63; ++round) {
      if (tid < 32) {
        int p = topi[tid], q = boti[tid];
        if (p > q) { int t = p; p = q; q = t; }
        pp[tid] = p; qq[tid] = q;
        float apq = A[p * AP_ + q];
        float app = A[p * AP_ + p], aqq = A[q * AP_ + q];
        float c = 1.f, s = 0.f;
        if (fabsf(apq) > 1e-20f) {
          float tau = (aqq - app) / (2.f * apq);
          float t2 = (tau >= 0.f ? 1.f : -1.f) / (fabsf(tau) + sqrtf(1.f + tau * tau));
          c = rsqrtf(1.f + t2 * t2);
          s = t2 * c;
        }
        rc[tid] = c; rs[tid] = s;
      }
      __syncthreads();
      {  // row update A = J^T A (thread = column index, pairs disjoint)
        const int kcol = tid;
        for (int j = 0; j < 32; ++j) {
          int p = pp[j], q = qq[j];
          float c = rc[j], s = rs[j];
          float ap = A[p * AP_ + kcol], aq = A[q * AP_ + kcol];
          A[p * AP_ + kcol] = c * ap - s * aq;
          A[q * AP_ + kcol] = s * ap + c * aq;
        }
      }
      __syncthreads();
      {  // column update A = A J, and V = V J (thread = row index)
        const int kr = tid;
        for (int j = 0; j < 32; ++j) {
          int p = pp[j], q = qq[j];
          float c = rc[j], s = rs[j];
          float ap = A[kr * AP_ + p], aq = A[kr * AP_ + q];
          A[kr * AP_ + p] = c * ap - s * aq;
          A[kr * AP_ + q] = s * ap + c * aq;
          float vp = V[kr * AP_ + p], vq = V[kr * AP_ + q];
          V[kr * AP_ + p] = c * vp - s * vq;
          V[kr * AP_ + q] = s * vp + c * vq;
        }
      }
      __syncthreads();
      if (tid == 0) {  // round-robin tournament rotation (top[0] fixed)
        int tmp = topi[31];
        for (int i2 = 31; i2 >= 2; --i2) topi[i2] = topi[i2 - 1];
        topi[1] = boti[0];
        for (int i2 = 0; i2 < 31; ++i2) boti[i2] = boti[i2 + 1];
        boti[31] = tmp;
      }
      __syncthreads();
    }
  }

  lw[tid] = logf(fmaxf(A[tid * AP_ + tid], 1e-30f));
  __syncthreads();  // lw captured; A is now reusable as logC output

  // logC = (V diag(lw)) V^T : 16 tiles of 16x16, 8 per wave, f32 WMMA k=4
  const int mrow = lane & 15;
  const int koff = (lane < 16) ? 0 : 2;
  for (int p = 0; p < 8; ++p) {
    const int tile = wave * 8 + p;
    const int ti = tile >> 2, tj = tile & 3;
    v8f acc = {};
    for (int k0 = 0; k0 < C_; k0 += 4) {
      const int kk = k0 + koff;
      v2f a  = {V[(ti * 16 + mrow) * AP_ + kk] * lw[kk],
                V[(ti * 16 + mrow) * AP_ + kk + 1] * lw[kk + 1]};
      v2f bb = {V[(tj * 16 + mrow) * AP_ + kk],
                V[(tj * 16 + mrow) * AP_ + kk + 1]};
      acc = __builtin_amdgcn_wmma_f32_16x16x4_f32(false, a, false, bb, (short)0, acc, false, false);
    }
    const int mbase = (lane < 16) ? 0 : 8;
    const int n = lane & 15;
#pragma unroll
    for (int r = 0; r < 8; ++r)
      A[(ti * 16 + mbase + r) * AP_ + tj * 16 + n] = acc[r];
  }
  __syncthreads();

  // upper-triangle vectorization with sqrt(2) off-diagonal scaling
  const float SQ2 = 1.41421356237f;
  float* vg = vecbuf + (size_t)g * VEC_;
  for (int idx = tid; idx < VEC_; idx += 64) {
    float fi = (129.f - sqrtf(16641.f - 8.f * (float)idx)) * 0.5f;
    int i = (int)fi;
    if (i > 63) i = 63;
    if (i < 0) i = 0;
    while (i > 0 && (64 * i - (i * (i - 1)) / 2) > idx) --i;
    while (i < 63 && (64 * (i + 1) - ((i + 1) * i) / 2) <= idx) ++i;
    int off = 64 * i - (i * (i - 1)) / 2;
    int j = i + (idx - off);
    float val = A[i * AP_ + j];
    vg[idx] = (i == j) ? val : val * SQ2;
  }
}

// ---------------------------------------------------------------------------
// K4a: Riemannian BN over (B,F) per tangent-vector element, in place.
// ---------------------------------------------------------------------------
__global__ void k_brn(float* __restrict__ vec, const float* __restrict__ brn_w) {
  const int v = blockIdx.x * blockDim.x + threadIdx.x;
  if (v >= VEC_) return;
  float s = 0.f, ss = 0.f;
  for (int g = 0; g < B_ * F_; ++g) {
    float x = vec[(size_t)g * VEC_ + v];
    s += x; ss += x * x;
  }
  const float m = s * (1.f / (B_ * F_));
  const float var = ss * (1.f / (B_ * F_)) - m * m;
  const float sc = rsqrtf(fmaxf(var, 0.f) + EPS_) * brn_w[v];
  for (int g = 0; g < B_ * F_; ++g) {
    size_t o = (size_t)g * VEC_ + v;
    vec[o] = (vec[o] - m) * sc;
  }
}

// ---------------------------------------------------------------------------
// K4b: BN1 over batch per feature (vec is exactly h[B][FEAT] flat), in place.
// ---------------------------------------------------------------------------
__global__ void k_bn1(float* __restrict__ h, const float* __restrict__ g1,
                      const float* __restrict__ b1v) {
  const int j = blockIdx.x * blockDim.x + threadIdx.x;
  if (j >= FEAT_) return;
  float s = 0.f, ss = 0.f;
  for (int b = 0; b < B_; ++b) {
    float x = h[(size_t)b * FEAT_ + j];
    s += x; ss += x * x;
  }
  const float m = s * (1.f / B_);
  const float var = ss * (1.f / B_) - m * m;
  const float sc = rsqrtf(fmaxf(var, 0.f) + EPS_);
  for (int b = 0; b < B_; ++b) {
    size_t o = (size_t)b * FEAT_ + j;
    h[o] = (h[o] - m) * sc * g1[j] + b1v[j];
  }
}

// ---------------------------------------------------------------------------
// K4c: h1[b][h] = gelu(h . W1[h] + b1[h]); one block per (b,h) = 256 blocks.
// ---------------------------------------------------------------------------
__global__ __launch_bounds__(256) void k_fc1(const float* __restrict__ h,
                                             const float* __restrict__ W1,
                                             const float* __restrict__ b1,
                                             float* __restrict__ h1) {
  const int bh = blockIdx.x;
  const int b = bh >> 3, hh = bh & 7;
  const int tid = threadIdx.x;
  __shared__ float red[256];
  float s = 0.f;
  const float* hr = h + (size_t)b * FEAT_;
  const float* wr = W1 + (size_t)hh * FEAT_;
  for (int j = tid; j < FEAT_; j += 256) s = fmaf(hr[j], wr[j], s);
  red[tid] = s;
  __syncthreads();
  for (int st = 128; st > 0; st >>= 1) {
    if (tid < st) red[tid] += red[tid + st];
    __syncthreads();
  }
  if (tid == 0) {
    float x = red[0] + b1[hh];
    h1[b * HID_ + hh] = 0.5f * x * (1.f + erff(x * 0.70710678118f));  // exact gelu
  }
}

// ---------------------------------------------------------------------------
// K4d: BN2 over batch + final linear head -> out[32][2].  Single block.
// ---------------------------------------------------------------------------
__global__ void k_head(const float* __restrict__ h1, const float* __restrict__ g2,
                       const float* __restrict__ bb2, const float* __restrict__ W2,
                       const float* __restrict__ b2, float* __restrict__ out) {
  __shared__ float m[HID_], iv[HID_];
  const int tid = threadIdx.x;
  if (tid < HID_) {
    float s = 0.f, ss = 0.f;
    for (int b = 0; b < B_; ++b) {
      float x = h1[b * HID_ + tid];
      s += x; ss += x * x;
    }
    float mm = s * (1.f / B_);
    float var = ss * (1.f / B_) - mm * mm;
    m[tid] = mm;
    iv[tid] = rsqrtf(fmaxf(var, 0.f) + EPS_);
  }
  __syncthreads();
  if (tid < B_ * NOUT_) {
    int b = tid >> 1, o = tid & 1;
    float acc = b2[o];
    for (int hh = 0; hh < HID_; ++hh) {
      float hn = (h1[b * HID_ + hh] - m[hh]) * iv[hh] * g2[hh] + bb2[hh];
      acc = fmaf(hn, W2[o * HID_ + hh], acc);
    }
    out[b * NOUT_ + o] = acc;
  }
}

// ---------------------------------------------------------------------------
extern "C" void kernel_launch(void* const* d_in, const int* in_sizes, int n_in,
                              void* d_out, int out_size, void* d_ws, size_t ws_size,
                              hipStream_t stream) {
  const float* X      = (const float*)d_in[0];
  const float* foi    = (const float*)d_in[1];
  const float* fwhm   = (const float*)d_in[2];
  const float* shrink = (const float*)d_in[3];
  const float* brn_w  = (const float*)d_in[4];
  const float* bn1_g  = (const float*)d_in[5];
  const float* bn1_b  = (const float*)d_in[6];
  const float* W1     = (const float*)d_in[7];
  const float* b1     = (const float*)d_in[8];
  const float* bn2_g  = (const float*)d_in[9];
  const float* bn2_b  = (const float*)d_in[10];
  const float* W2     = (const float*)d_in[11];
  const float* b2     = (const float*)d_in[12];

  float* w    = (float*)d_ws;
  float* kc   = w;                                   // F*K      = 1250
  float* ks   = kc + F_ * K_;                        // F*K      = 1250
  float* Craw = ks + F_ * K_;                        // 320*4096 = 1,310,720
  float* vec  = Craw + (size_t)B_ * F_ * C_ * C_;    // 320*2080 = 665,600 (== h[B][FEAT])
  float* h1   = vec + (size_t)B_ * FEAT_;            // 256
  // total ~7.9 MB of workspace

  k_build<<<F_, 128, 0, stream>>>(foi, fwhm, kc, ks);
  k_conv_cov<<<B_ * F_, 256, 0, stream>>>(X, kc, ks, Craw);
  k_eig_logm<<<B_ * F_, 64, 0, stream>>>(Craw, shrink, vec);
  k_brn<<<(VEC_ + 255) / 256, 256, 0, stream>>>(vec, brn_w);
  k_bn1<<<(FEAT_ + 255) / 256, 256, 0, stream>>>(vec, bn1_g, bn1_b);
  k_fc1<<<B_ * HID_, 256, 0, stream>>>(vec, W1, b1, h1);
  k_head<<<1, 64, 0, stream>>>(h1, bn2_g, bn2_b, W2, b2, (float*)d_out);
}